// HeteroGraphSage_66365834658212
// MI455X (gfx1250) — compile-verified
//
#include <hip/hip_runtime.h>

typedef float v2f __attribute__((ext_vector_type(2)));
typedef float v8f __attribute__((ext_vector_type(8)));

constexpr int N_NODES = 50000;
constexpr int N_REL   = 3;
constexpr int N_EDGES = 500000;
constexpr int D_IN  = 128;
constexpr int D_HID = 256;
constexpr int D_OUT = 128;

// ---------------------------------------------------------------------------
// Sum 3 relation slices of a weight/bias tensor: out[i] = sum_r w[r*sz + i]
// ---------------------------------------------------------------------------
__global__ void relsum_kernel(const float* __restrict__ w, float* __restrict__ out, int sz) {
    int i = blockIdx.x * blockDim.x + threadIdx.x;
    if (i < sz) out[i] = w[i] + w[sz + i] + w[2 * sz + i];
}

// ---------------------------------------------------------------------------
// Per-relation in-degree via f32 atomics (matches segment_sum of ones).
// ---------------------------------------------------------------------------
__global__ void degree_kernel(const int* __restrict__ dst, float* __restrict__ deg) {
    int i = blockIdx.x * blockDim.x + threadIdx.x;
    if (i < N_REL * N_EDGES) {
        int r = i / N_EDGES;
        atomicAdd(deg + (long long)r * N_NODES + dst[i], 1.0f);
    }
}

// ---------------------------------------------------------------------------
// One wave32 per (rel, edge): src/dst forced into SGPRs (readfirstlane), lanes
// gather the feature row as coalesced float4 and scatter with
// global_atomic_add_f32.  Accumulators ([3,50K,D] = 77/154 MB) fit in the
// 192 MB L2, so the atomic traffic resolves at L2 bandwidth.
// ---------------------------------------------------------------------------
__global__ void scatter_add_kernel(const float* __restrict__ x,
                                   const int* __restrict__ src,
                                   const int* __restrict__ dst,
                                   float* __restrict__ sum, int D) {
    long long w = (long long)blockIdx.x * (blockDim.x >> 5) + (threadIdx.x >> 5);
    if (w >= (long long)N_REL * N_EDGES) return;
    const int lane = threadIdx.x & 31;
    int r = (int)(w / N_EDGES);
    int s = __builtin_amdgcn_readfirstlane(src[w]);
    int d = __builtin_amdgcn_readfirstlane(dst[w]);
    const float* xrow = x + (long long)s * D;
    float*       orow = sum + ((long long)r * N_NODES + d) * D;
    for (int c = lane * 4; c < D; c += 128) {
        float4 v = *(const float4*)(xrow + c);
        atomicAdd(orow + c + 0, v.x);
        atomicAdd(orow + c + 1, v.y);
        atomicAdd(orow + c + 2, v.z);
        atomicAdd(orow + c + 3, v.w);
    }
}

// ---------------------------------------------------------------------------
// Fused multi-segment GEMM with V_WMMA_F32_16X16X4_F32 (fp32, exact vs ref):
//   out = act( sum_seg (A[seg] * rowscale[seg]) @ B[seg] + bias )
// A[seg]: [M,K] row-major.  B[seg]: [K,N] row-major.  rowscale folds the
// SAGE mean (1/max(deg,1)) into the A-read, eliminating a separate
// normalization pass over the 77/154 MB accumulators.
//
// Block = 128 threads = 4 wave32s; each wave owns a 32(M) x 64(N) C tile
// (2x4 v8f accumulators).  B's 64-column slice is staged in LDS once per
// segment in a [K/4][64][half] interleaved layout so each lane's fragment is
// one conflict-free ds_load_b64.
//
// CDNA5 fragment layouts (ISA 7.12.2):
//   A 16x4 f32 (2 VGPR): lanes 0-15 row M=lane hold K={k,k+1};
//                        lanes 16-31 row M=lane-16 hold K={k+2,k+3}
//   B 4x16 f32 (2 VGPR): lanes 0-15 col N=lane hold K={k,k+1};
//                        lanes 16-31 col N=lane-16 hold K={k+2,k+3}
//   C 16x16 f32 (8 VGPR): VGPR i = rows {i, i+8} x col lane%16
// ---------------------------------------------------------------------------
struct GemmSegs {
    const float* A[4];
    const float* B[4];
    const float* deg[4];   // nullptr => self path (no mean scaling)
};

template <int K, bool RELU>
__global__ __launch_bounds__(128)
void sage_wmma_gemm_kernel(GemmSegs segs, int M, int N,
                           const float* __restrict__ bias,
                           float* __restrict__ out) {
    __shared__ v2f bsh[(K / 4) * 128];   // [K/4][64 cols][2 halves] = K*256 bytes

    const int lane    = threadIdx.x & 31;
    const int wave    = threadIdx.x >> 5;
    const int halfsel = lane >> 4;       // 0: lanes 0-15, 1: lanes 16-31
    const int lcol    = lane & 15;
    const int kk      = halfsel * 2;     // K sub-offset held by this half-wave

    const int rowW = blockIdx.x * 128 + wave * 32;   // this wave's first row
    const int col0 = blockIdx.y * 64;                // this block's first col

    v8f acc[2][4];
#pragma unroll
    for (int mi = 0; mi < 2; ++mi)
#pragma unroll
        for (int ni = 0; ni < 4; ++ni)
            acc[mi][ni] = (v8f){};

    // Clamped A row per M sub-tile (tail rows load row M-1; stores guarded).
    int rA[2];
#pragma unroll
    for (int mi = 0; mi < 2; ++mi) {
        int r = rowW + mi * 16 + lcol;
        rA[mi] = r < M ? r : (M - 1);
    }

    for (int seg = 0; seg < 4; ++seg) {
        const float* __restrict__ A  = segs.A[seg];
        const float* __restrict__ B  = segs.B[seg];
        const float* __restrict__ dg = segs.deg[seg];

        // ---- Stage this segment's B slice [K x 64] into LDS ----
        __syncthreads();   // previous segment's reads done
#pragma unroll 4
        for (int idx = threadIdx.x; idx < (K / 4) * 128; idx += 128) {
            int kg = idx >> 7;        // K group of 4
            int h  = (idx >> 6) & 1;  // half (which K pair)
            int c  = idx & 63;        // column within tile
            const float* bp = B + (long long)(4 * kg + 2 * h) * N + col0 + c;
            v2f v;
            v.x = bp[0];
            v.y = bp[N];
            bsh[(kg * 64 + c) * 2 + h] = v;
        }
        __syncthreads();

        // Mean scale folded into the A operand (uniform scalar branch).
        float sc0 = 1.0f, sc1 = 1.0f;
        if (dg) {
            sc0 = 1.0f / fmaxf(dg[rA[0]], 1.0f);
            sc1 = 1.0f / fmaxf(dg[rA[1]], 1.0f);
        }

        for (int k = 0; k < K; k += 4) {
            const int kg = k >> 2;
            v2f bfrag[4];
#pragma unroll
            for (int ni = 0; ni < 4; ++ni)
                bfrag[ni] = bsh[(kg * 64 + ni * 16 + lcol) * 2 + halfsel];

            v2f a0 = *(const v2f*)(A + (long long)rA[0] * K + k + kk);
            v2f a1 = *(const v2f*)(A + (long long)rA[1] * K + k + kk);
            a0 *= sc0;
            a1 *= sc1;

#pragma unroll
            for (int ni = 0; ni < 4; ++ni)
                acc[0][ni] = __builtin_amdgcn_wmma_f32_16x16x4_f32(
                    false, a0, false, bfrag[ni], (short)0, acc[0][ni], false, false);
#pragma unroll
            for (int ni = 0; ni < 4; ++ni)
                acc[1][ni] = __builtin_amdgcn_wmma_f32_16x16x4_f32(
                    false, a1, false, bfrag[ni], (short)0, acc[1][ni], false, false);
        }
    }

    // Epilogue: bias (+ReLU) and store.
    float bv[4];
#pragma unroll
    for (int ni = 0; ni < 4; ++ni) bv[ni] = bias[col0 + ni * 16 + lcol];

#pragma unroll
    for (int mi = 0; mi < 2; ++mi) {
#pragma unroll
        for (int i = 0; i < 8; ++i) {
            int row = rowW + mi * 16 + halfsel * 8 + i;
            if (row < M) {
#pragma unroll
                for (int ni = 0; ni < 4; ++ni) {
                    float v = acc[mi][ni][i] + bv[ni];
                    if (RELU) v = fmaxf(v, 0.0f);
                    out[(long long)row * N + col0 + ni * 16 + lcol] = v;
                }
            }
        }
    }
}

// ---------------------------------------------------------------------------
// Host-side pipeline
// ---------------------------------------------------------------------------
static inline int nblk(long long total, int per) { return (int)((total + per - 1) / per); }

extern "C" void kernel_launch(void* const* d_in, const int* in_sizes, int n_in,
                              void* d_out, int out_size, void* d_ws, size_t ws_size,
                              hipStream_t stream) {
    (void)in_sizes; (void)n_in; (void)out_size; (void)ws_size;

    const float* x        = (const float*)d_in[0];
    const int*   src      = (const int*)d_in[1];
    const int*   dst      = (const int*)d_in[2];
    const float* w_self1  = (const float*)d_in[3];
    const float* w_neigh1 = (const float*)d_in[4];
    const float* b1       = (const float*)d_in[5];
    const float* w_self2  = (const float*)d_in[6];
    const float* w_neigh2 = (const float*)d_in[7];
    const float* b2       = (const float*)d_in[8];
    float*       out      = (float*)d_out;

    // Workspace carve-out (256B aligned slices).
    char*  ws  = (char*)d_ws;
    size_t off = 0;
    auto carve = [&](size_t bytes) -> float* {
        float* p = (float*)(ws + off);
        off = (off + bytes + 255) & ~(size_t)255;
        return p;
    };
    float* deg   = carve((size_t)N_REL * N_NODES * 4);            //  0.6 MB
    float* hn1   = carve((size_t)N_REL * N_NODES * D_IN  * 4);    // 76.8 MB (raw sums)
    float* h     = carve((size_t)N_NODES * D_HID * 4);            // 51.2 MB
    float* hn2   = carve((size_t)N_REL * N_NODES * D_HID * 4);    // 153.6 MB (raw sums)
    float* wsum1 = carve((size_t)D_IN * D_HID * 4);
    float* bsum1 = carve((size_t)D_HID * 4);
    float* wsum2 = carve((size_t)D_HID * D_OUT * 4);
    float* bsum2 = carve((size_t)D_OUT * 4);

    // Zero the atomic accumulators (graph-capture-safe memset nodes).
    hipMemsetAsync(deg, 0, (size_t)N_REL * N_NODES * 4, stream);
    hipMemsetAsync(hn1, 0, (size_t)N_REL * N_NODES * D_IN  * 4, stream);
    hipMemsetAsync(hn2, 0, (size_t)N_REL * N_NODES * D_HID * 4, stream);

    // Fold the relation sum of the self-path into single weights/biases:
    //   sum_r (x @ Ws_r + b_r) = x @ (sum_r Ws_r) + sum_r b_r
    relsum_kernel<<<nblk(D_IN * D_HID, 256), 256, 0, stream>>>(w_self1, wsum1, D_IN * D_HID);
    relsum_kernel<<<1, 256, 0, stream>>>(b1, bsum1, D_HID);
    relsum_kernel<<<nblk(D_HID * D_OUT, 256), 256, 0, stream>>>(w_self2, wsum2, D_HID * D_OUT);
    relsum_kernel<<<1, 128, 0, stream>>>(b2, bsum2, D_OUT);

    // Degrees (shared by both layers).
    degree_kernel<<<nblk((long long)N_REL * N_EDGES, 256), 256, 0, stream>>>(dst, deg);

    // ---- Layer 1 aggregation: neighbor sums of x (one wave per edge) ----
    scatter_add_kernel<<<nblk((long long)N_REL * N_EDGES, 8), 256, 0, stream>>>(
        x, src, dst, hn1, D_IN);

    // ---- Layer 1 GEMM: h = relu(x@Wsum1 + sum_r (hn1_r/deg_r)@Wn1_r + bsum1) ----
    {
        GemmSegs segs;
        segs.A[0] = x;      segs.B[0] = wsum1;  segs.deg[0] = nullptr;
        for (int r = 0; r < N_REL; ++r) {
            segs.A[r + 1]   = hn1 + (size_t)r * N_NODES * D_IN;
            segs.B[r + 1]   = w_neigh1 + (size_t)r * D_IN * D_HID;
            segs.deg[r + 1] = deg + (size_t)r * N_NODES;
        }
        dim3 grid(nblk(N_NODES, 128), D_HID / 64);
        sage_wmma_gemm_kernel<D_IN, true><<<grid, 128, 0, stream>>>(
            segs, N_NODES, D_HID, bsum1, h);
    }

    // ---- Layer 2 aggregation: neighbor sums of h ----
    scatter_add_kernel<<<nblk((long long)N_REL * N_EDGES, 8), 256, 0, stream>>>(
        h, src, dst, hn2, D_HID);

    // ---- Layer 2 GEMM: out = h@Wsum2 + sum_r (hn2_r/deg_r)@Wn2_r + bsum2 ----
    {
        GemmSegs segs;
        segs.A[0] = h;      segs.B[0] = wsum2;  segs.deg[0] = nullptr;
        for (int r = 0; r < N_REL; ++r) {
            segs.A[r + 1]   = hn2 + (size_t)r * N_NODES * D_HID;
            segs.B[r + 1]   = w_neigh2 + (size_t)r * D_HID * D_OUT;
            segs.deg[r + 1] = deg + (size_t)r * N_NODES;
        }
        dim3 grid(nblk(N_NODES, 128), D_OUT / 64);
        sage_wmma_gemm_kernel<D_HID, false><<<grid, 128, 0, stream>>>(
            segs, N_NODES, D_OUT, bsum2, out);
    }
}